// SegTerm_70248485093641
// MI455X (gfx1250) — compile-verified
//
#include <hip/hip_runtime.h>

// Problem constants (from the reference)
#define HDIM 256
#define WDIM 512
#define HW   (HDIM * WDIM)          // 131072 pixels per plane
#define NUM_SEG_CLASSES 19
#define NUM_STUFF 11                // 19 - 8
#define NUM_BOXES 128
#define BOX_SCALE 0.25f

// ---------------------------------------------------------------------------
// Kernel 1: seg_stuff_energy = seg_score[:, :11]  (pure 5.5 MB copy)
// float4 path -> global_load_b128 / global_store_b128; one prefetch ahead
// per thread to keep the load pipe full (global_prefetch_b8 on gfx1250).
// ---------------------------------------------------------------------------
__global__ __launch_bounds__(256) void stuff_copy_kernel(
    const float* __restrict__ src, float* __restrict__ dst)
{
    const int i = blockIdx.x * 256 + threadIdx.x;     // float4 index
    const float4* __restrict__ s = (const float4*)src;
    float4*       __restrict__ d = (float4*)dst;
    // grid is sized exactly: NUM_STUFF*HW/4 / 256 blocks
    __builtin_prefetch((const void*)(s + i + 4096), 0, 0); // speculative, next ~64KB
    d[i] = s[i];
}

// ---------------------------------------------------------------------------
// Kernel 2: seg_inst_energy[n, y, x] =
//     (cls[n]!=0 && y0<=y<y1 && x0<=x<x1) ? seg_score[map(cls[n]), y, x] : 0
//
// blockIdx.y = box index n  -> cls/box params are wave-uniform (SALU/s_load).
// Each thread owns a 4-pixel (float4) group in one row (W=512 is /4).
// Groups fully outside the rectangle store zeros without reading the source,
// so source reads are limited to the masked area (~2-3% of pixels).
// ---------------------------------------------------------------------------
__global__ __launch_bounds__(256) void inst_energy_kernel(
    const int*   __restrict__ cls_indices,
    const float* __restrict__ seg_score,
    const float* __restrict__ boxes,
    float*       __restrict__ out)
{
    const int n   = blockIdx.y;                       // box id, uniform per block
    const int g   = blockIdx.x * 256 + threadIdx.x;   // float4 group in plane
    const int pix = g << 2;                           // first pixel of group
    const int y   = pix >> 9;                         // / 512
    const int xb  = pix & (WDIM - 1);                 // % 512

    // Uniform per-block scalars (compiler lowers to s_load + SALU)
    const int   cls = cls_indices[n];
    const float bx0 = boxes[n * 5 + 1] * BOX_SCALE;
    const float by0 = boxes[n * 5 + 2] * BOX_SCALE;
    const float bx1 = boxes[n * 5 + 3] * BOX_SCALE;
    const float by1 = boxes[n * 5 + 4] * BOX_SCALE;

    const int x0 = (int)floorf(bx0);
    const int y0 = (int)floorf(by0);
    const int x1 = __float2int_rn(bx1) + 1;           // RTE == jnp.round
    const int y1 = __float2int_rn(by1) + 1;

    int mapped = cls + 10;                            // clip(cls+10, 0, 18)
    mapped = mapped > (NUM_SEG_CLASSES - 1) ? (NUM_SEG_CLASSES - 1) : mapped;
    mapped = mapped < 0 ? 0 : mapped;

    float4 v = make_float4(0.f, 0.f, 0.f, 0.f);
    if (cls != 0 && y >= y0 && y < y1 && (xb + 4) > x0 && xb < x1) {
        // whole group's row segment is in-bounds of the source channel plane
        const float4 s = *(const float4*)(seg_score + (size_t)mapped * HW + pix);
        v.x = (xb + 0 >= x0 && xb + 0 < x1) ? s.x : 0.f;
        v.y = (xb + 1 >= x0 && xb + 1 < x1) ? s.y : 0.f;
        v.z = (xb + 2 >= x0 && xb + 2 < x1) ? s.z : 0.f;
        v.w = (xb + 3 >= x0 && xb + 3 < x1) ? s.w : 0.f;
    }
    *(float4*)(out + (size_t)n * HW + pix) = v;       // global_store_b128
}

// ---------------------------------------------------------------------------
// Launch. Output layout (flat, reference return order):
//   [0, 11*HW)            : seg_stuff_energy
//   [11*HW, 11*HW+128*HW) : seg_inst_energy
// ---------------------------------------------------------------------------
extern "C" void kernel_launch(void* const* d_in, const int* in_sizes, int n_in,
                              void* d_out, int out_size, void* d_ws, size_t ws_size,
                              hipStream_t stream)
{
    const int*   cls   = (const int*)  d_in[0];   // (128,) int32
    const float* seg   = (const float*)d_in[1];   // (1,19,256,512) f32
    const float* boxes = (const float*)d_in[2];   // (128,5) f32
    float*       out   = (float*)d_out;

    // Stuff copy: 11*HW/4 = 360448 float4 -> exactly 1408 blocks of 256
    const int n4 = NUM_STUFF * HW / 4;
    stuff_copy_kernel<<<dim3(n4 / 256), dim3(256), 0, stream>>>(seg, out);

    // Inst energy: 128 planes x (HW/4 = 32768 groups -> 128 blocks of 256)
    inst_energy_kernel<<<dim3(HW / 4 / 256, NUM_BOXES), dim3(256), 0, stream>>>(
        cls, seg, boxes, out + (size_t)NUM_STUFF * HW);
}